// scPRS_37460704755979
// MI455X (gfx1250) — compile-verified
//
#include <hip/hip_runtime.h>

typedef __attribute__((ext_vector_type(2))) float v2f;
typedef __attribute__((ext_vector_type(8))) float v8f;

// ---------------------------------------------------------------------------
// Kernel 1: h[n*B+b] = (sum_d x[b,n,d]*|w[d]|)/D + bias1
// One wave32 per 16-row tile (16 consecutive n, fixed b), using
// V_WMMA_F32_16X16X4_F32 with B-matrix = |w| broadcast across 16 columns.
// Hand double-buffered: next 32-k chunk is loaded before current chunk's
// 8 WMMAs issue, forcing staggered s_wait counts (loads stay in flight).
// ---------------------------------------------------------------------------
__device__ __forceinline__ void load_chunk(const float* __restrict__ rowp,
                                           const float* __restrict__ lds_w,
                                           int kb, int koff,
                                           v2f* __restrict__ av,
                                           v2f* __restrict__ wv)
{
#pragma unroll
    for (int j = 0; j < 8; ++j) {
        const int idx = kb + 4 * j + koff;
        av[j].x = rowp[idx];
        av[j].y = rowp[idx + 1];
        wv[j].x = lds_w[idx];
        wv[j].y = lds_w[idx + 1];
    }
}

__global__ __launch_bounds__(256) void k_dot_wmma(
    const float* __restrict__ x, const float* __restrict__ param,
    const float* __restrict__ bias1, float* __restrict__ h,
    int N, int D, int B, int tilesTotal)
{
    extern __shared__ float lds_w[];   // D floats of |param|
    for (int i = threadIdx.x; i < D; i += blockDim.x)
        lds_w[i] = fabsf(param[i]);
    __syncthreads();

    const int wave = threadIdx.x >> 5;
    const int lane = threadIdx.x & 31;
    int tile = blockIdx.x * (blockDim.x >> 5) + wave;
    const bool valid = (tile < tilesTotal);   // wave-uniform
    if (!valid) tile = 0;                     // keep work well-defined; skip store

    const int tilesPerB = N >> 4;             // N % 16 == 0
    const int b  = tile / tilesPerB;
    const int n0 = (tile - b * tilesPerB) << 4;

    // 32-bit A-matrix 16x4 layout: lanes 0-15 -> K=0,1 ; lanes 16-31 -> K=2,3
    const int row  = lane & 15;               // M = 0..15
    const int koff = (lane >> 4) << 1;        // 0 or 2
    const float* rowp = x + ((size_t)b * N + (size_t)(n0 + row)) * (size_t)D;

    v8f c = {};
    int k = 0;
    const int kmain = D & ~31;                // full 32-k chunks

    if (kmain > 0) {
        v2f a[8], w8[8];
        load_chunk(rowp, lds_w, 0, koff, a, w8);      // prologue: chunk 0
        for (k = 32; k < kmain; k += 32) {
            __builtin_prefetch(rowp + k + 224, 0, 3); // ~1KB ahead, one per line pair
            v2f an[8], wn[8];
            load_chunk(rowp, lds_w, k, koff, an, wn); // issue next chunk's loads
#pragma unroll
            for (int j = 0; j < 8; ++j)               // compute current chunk
                c = __builtin_amdgcn_wmma_f32_16x16x4_f32(false, a[j], false, w8[j],
                                                          (short)0, c, false, false);
#pragma unroll
            for (int j = 0; j < 8; ++j) { a[j] = an[j]; w8[j] = wn[j]; }
        }
#pragma unroll
        for (int j = 0; j < 8; ++j)                   // epilogue: last full chunk
            c = __builtin_amdgcn_wmma_f32_16x16x4_f32(false, a[j], false, w8[j],
                                                      (short)0, c, false, false);
        k = kmain;
    }
    for (; k < D; k += 4) {                           // tail (D % 32)
        v2f a, bm;
        a.x  = rowp[k + koff];
        a.y  = rowp[k + koff + 1];
        bm.x = lds_w[k + koff];
        bm.y = lds_w[k + koff + 1];
        c = __builtin_amdgcn_wmma_f32_16x16x4_f32(false, a, false, bm,
                                                  (short)0, c, false, false);
    }

    const float invD = 1.0f / (float)D;
    const float bb   = bias1[0];
    // C/D layout: lane l, VGPR v -> M = v + 8*(l>=16), N = l%16 (all cols equal)
    if (valid && ((lane & 15) < 8)) {
        const int s = lane & 7;
        float val = c[0];                             // select chain, no scratch
#pragma unroll
        for (int j = 1; j < 8; ++j) val = (s == j) ? c[j] : val;
        int m = s + ((lane >> 4) << 3);
        h[(size_t)(n0 + m) * B + b] = val * invD + bb;
    }
}

// ---------------------------------------------------------------------------
// Helpers for graph aggregation
// ---------------------------------------------------------------------------
__global__ void k_zero(float* __restrict__ p, int n) {
    int i = blockIdx.x * blockDim.x + threadIdx.x;
    if (i < n) p[i] = 0.0f;
}

__global__ void k_degree(const int* __restrict__ dst, float* __restrict__ cnt, int E) {
    int e = blockIdx.x * blockDim.x + threadIdx.x;
    if (e < E) atomicAdd(&cnt[dst[e]], 1.0f);
}

__global__ void k_invcnt(const float* __restrict__ cnt, float* __restrict__ inv, int N) {
    int n = blockIdx.x * blockDim.x + threadIdx.x;
    if (n < N) inv[n] = 1.0f / fmaxf(cnt[n], 1.0f);
}

// msg = |p1|*h[src] + |p2|*h[dst] scatter-added into agg[dst]
__global__ void k_scatter(const int* __restrict__ src, const int* __restrict__ dst,
                          const float* __restrict__ hin, float* __restrict__ agg,
                          const float* __restrict__ p1, const float* __restrict__ p2,
                          int E, int B)
{
    int t = blockIdx.x * blockDim.x + threadIdx.x;
    if (t >= E * B) return;
    int e = t >> 4;            // B == 16
    int b = t & 15;
    int s = src[e], d = dst[e];
    float ap1 = fabsf(p1[0]);
    float ap2 = fabsf(p2[0]);
    float val = ap1 * hin[(size_t)s * B + b] + ap2 * hin[(size_t)d * B + b];
    atomicAdd(&agg[(size_t)d * B + b], val);
}

__global__ void k_finalize(const float* __restrict__ agg, const float* __restrict__ inv,
                           const float* __restrict__ bias, float* __restrict__ hout,
                           int N, int B)
{
    int i = blockIdx.x * blockDim.x + threadIdx.x;
    if (i >= N * B) return;
    int n = i / B;
    float a = agg[i] * inv[n] + bias[0];
    hout[i] = (a > 0.0f) ? a : 0.1f * a;   // leaky_relu(0.1)
}

// out[b] = sum_n h[n*B+b]*w[n] + pred_b
__global__ __launch_bounds__(256) void k_pred(const float* __restrict__ h,
                                              const float* __restrict__ w,
                                              const float* __restrict__ pb,
                                              float* __restrict__ out, int N, int B)
{
    __shared__ float red[256];
    int b = blockIdx.x;
    float s = 0.0f;
    for (int n = threadIdx.x; n < N; n += blockDim.x)
        s += h[(size_t)n * B + b] * w[n];
    red[threadIdx.x] = s;
    __syncthreads();
    for (int off = 128; off > 0; off >>= 1) {
        if ((int)threadIdx.x < off) red[threadIdx.x] += red[threadIdx.x + off];
        __syncthreads();
    }
    if (threadIdx.x == 0) out[b] = red[0] + pb[0];
}

// ---------------------------------------------------------------------------
extern "C" void kernel_launch(void* const* d_in, const int* in_sizes, int n_in,
                              void* d_out, int out_size, void* d_ws, size_t ws_size,
                              hipStream_t stream)
{
    const float* x        = (const float*)d_in[0];
    const int*   edge     = (const int*)  d_in[1];
    /* edge_weight d_in[2] unused, as in reference */
    const float* param    = (const float*)d_in[3];
    const float* bias1    = (const float*)d_in[4];
    const float* p1       = (const float*)d_in[5];
    const float* p2       = (const float*)d_in[6];
    const float* agg_bias = (const float*)d_in[7];
    const float* pred_w   = (const float*)d_in[8];
    const float* pred_b   = (const float*)d_in[9];

    const int D  = in_sizes[3];                 // parameter: D x 1
    const int E  = in_sizes[2];                 // edge_weight: E
    const int N  = in_sizes[8];                 // pred_w: 1 x N
    const int B  = in_sizes[0] / N / D;         // x: B x N x D
    const int NG = in_sizes[5];                 // p1: NG x 1 x 1

    const int* src = edge;
    const int* dst = edge + E;

    // workspace carve-out (floats)
    float* ws  = (float*)d_ws;
    float* h0  = ws;
    float* h1  = h0  + (size_t)N * B;
    float* agg = h1  + (size_t)N * B;
    float* cnt = agg + (size_t)N * B;
    float* inv = cnt + N;

    const int T = 256;

    // in-degree -> inverse counts
    k_zero  <<<(N + T - 1) / T, T, 0, stream>>>(cnt, N);
    k_degree<<<(E + T - 1) / T, T, 0, stream>>>(dst, cnt, E);
    k_invcnt<<<(N + T - 1) / T, T, 0, stream>>>(cnt, inv, N);

    // initial projection via WMMA
    const int tiles = B * (N >> 4);             // one wave32 per 16-row tile
    const int wavesPerBlk = T >> 5;
    k_dot_wmma<<<(tiles + wavesPerBlk - 1) / wavesPerBlk, T,
                 (size_t)D * sizeof(float), stream>>>(
        x, param, bias1, h0, N, D, B, tiles);

    // 3 message-passing layers, ping-pong h buffers
    float* hin = h0; float* hout = h1;
    const int NB = N * B;
    for (int i = 0; i < NG; ++i) {
        k_zero    <<<(NB + T - 1) / T, T, 0, stream>>>(agg, NB);
        k_scatter <<<((size_t)E * B + T - 1) / T, T, 0, stream>>>(
            src, dst, hin, agg, p1 + i, p2 + i, E, B);
        k_finalize<<<(NB + T - 1) / T, T, 0, stream>>>(
            agg, inv, agg_bias + i, hout, N, B);
        float* tmp = hin; hin = hout; hout = tmp;
    }

    // prediction head: out[b] = h[:,b] . pred_w + pred_b
    k_pred<<<B, T, 0, stream>>>(hin, pred_w, pred_b, (float*)d_out, N, B);
}